// Mix_78185584656633
// MI455X (gfx1250) — compile-verified
//
#include <hip/hip_runtime.h>
#include <hip/hip_bf16.h>
#include <math.h>

typedef __attribute__((ext_vector_type(2))) float v2f;
typedef __attribute__((ext_vector_type(8))) float v8f;

#define D_DIM 128
#define MAX_DEG 64

__device__ __forceinline__ unsigned pcg_hash(unsigned x) {
  x = x * 747796405u + 2891336453u;
  unsigned w = ((x >> ((x >> 28u) + 4u)) ^ x) * 277803737u;
  return (w >> 22u) ^ w;
}

// ---------------------------------------------------------------------------
// Projection GEMM via V_WMMA_F32_16X16X4_F32.
// Computes outA[r] = H[r,:] . attA  (and outB[r] = H[r,:] . attB if useB)
// as a (rows x 128) @ (128 x 16) GEMM where only B columns 0/1 are nonzero.
// One wave32 per 16-row tile; 32 WMMA steps cover K=128.
//
// A 16x4 f32 layout (ISA 7.12.2): lanes 0-15 = rows, VGPR0=K0,VGPR1=K1;
// lanes 16-31 = rows, VGPR0=K2,VGPR1=K3.
// B 4x16 f32 layout: lanes 0-15 = cols with K0(V0),K1(V1); lanes 16-31 K2,K3.
// C/D 16x16 f32: VGPR j -> M=j (lanes 0-15) / M=j+8 (lanes 16-31), N=lane&15.
//
// B fragment is built branch-free: unconditional att loads + per-lane
// multiplier masks -> v_cndmask/v_mul, no exec-mask ladders per WMMA step.
// ---------------------------------------------------------------------------
__global__ void proj_wmma_kernel(const float* __restrict__ H, int rows,
                                 const float* __restrict__ attA,
                                 const float* __restrict__ attB,  // valid ptr
                                 int useB,
                                 float* __restrict__ outA,
                                 float* __restrict__ outB) {
  const int lane   = threadIdx.x & 31;
  const int waveId = threadIdx.x >> 5;
  const int tile   = blockIdx.x * (blockDim.x >> 5) + waveId;
  const int tileRow = tile * 16;
  if (tileRow >= rows) return;                 // wave-uniform: EXEC stays full

  const int col  = lane & 15;                  // A: row-in-tile, B/D: column
  const int half = lane >> 4;                  // K sub-chunk selector

  int arow = tileRow + col;
  if (arow >= rows) arow = rows - 1;           // tail clamp (rows%16==0 here)
  const float* __restrict__ Arow = H + (size_t)arow * D_DIM;

  const float selA = (col == 0) ? 1.f : 0.f;
  const float selB = (col == 1 && useB) ? 1.f : 0.f;

  v8f c = {};
  #pragma unroll
  for (int k = 0; k < D_DIM; k += 4) {
    const int kk = k + 2 * half;
    v2f a, b;
    a.x = Arow[kk];
    a.y = Arow[kk + 1];
    const float a0 = attA[kk];
    const float a1 = attA[kk + 1];
    const float b0 = attB[kk];
    const float b1 = attB[kk + 1];
    b.x = selA * a0 + selB * b0;
    b.y = selA * a1 + selB * b1;
    c = __builtin_amdgcn_wmma_f32_16x16x4_f32(false, a, false, b,
                                              (short)0, c, false, false);
  }

  if (col == 0) {                              // column 0 = dot with attA
    #pragma unroll
    for (int j = 0; j < 8; ++j) {
      int r = tileRow + half * 8 + j;
      if (r < rows) outA[r] = c[j];
    }
  } else if (col == 1 && useB) {               // column 1 = dot with attB
    #pragma unroll
    for (int j = 0; j < 8; ++j) {
      int r = tileRow + half * 8 + j;
      if (r < rows) outB[r] = c[j];
    }
  }
}

// ---------------------------------------------------------------------------
// Sampling + softmax attention + weighted aggregation. One wave32 per node:
// lanes < S own one sample each (hash-based sampling, scalar projection
// gather, leaky_relu logit), wave-shuffle softmax across S lanes, then S
// coalesced float4 row gathers (lane i covers dims 4i..4i+3) with a one-step
// prefetch, residual init from h_self, float4 store.
// ---------------------------------------------------------------------------
__global__ void gat_agg_kernel(const float* __restrict__ h_self,
                               const float* __restrict__ h_nei,
                               const int* __restrict__ nei_idx,
                               const int* __restrict__ deg,
                               const float* __restrict__ projSelf,
                               const float* __restrict__ projNei,
                               float* __restrict__ out,
                               int n_nodes, int S, unsigned seed) {
  const int lane = threadIdx.x & 31;
  const int wave = threadIdx.x >> 5;
  const int n = blockIdx.x * (blockDim.x >> 5) + wave;
  if (n >= n_nodes) return;                    // wave-uniform

  int dgr = deg[n];
  if (dgr < 1) dgr = 1;
  const float psel = projSelf[n];

  float logit = -3.402823e38f;
  int idx = 0;
  if (lane < S) {
    unsigned h = pcg_hash((unsigned)n * 0x9E3779B9u ^
                          ((unsigned)lane * 0x85EBCA6Bu) ^ seed);
    float u = (float)(h & 0xFFFFFFu) * (1.0f / 16777216.0f);
    int pos = (int)(u * (float)dgr);
    if (pos > dgr - 1) pos = dgr - 1;
    idx = nei_idx[(size_t)n * MAX_DEG + pos];
    float pn = projNei[idx];
    float x = psel + pn;
    logit = (x > 0.f) ? x : 0.01f * x;         // leaky_relu, slope 0.01
  }

  // softmax across the S sample lanes (S = 8 or 16, power of two)
  float m = logit;
  for (int off = 1; off < S; off <<= 1) m = fmaxf(m, __shfl_xor(m, off, S));
  float e = (lane < S) ? __expf(logit - m) : 0.f;
  float ssum = e;
  for (int off = 1; off < S; off <<= 1) ssum += __shfl_xor(ssum, off, S);
  const float wgt = (lane < S) ? (e / ssum) : 0.f;

  // residual init + weighted neighbor gather (L2-resident tables)
  const float4* __restrict__ selfv =
      (const float4*)h_self + (size_t)n * (D_DIM / 4);
  float4 acc = selfv[lane];
  const float4* __restrict__ neiBase = (const float4*)h_nei;
  for (int s = 0; s < S; ++s) {
    const int   is = __shfl(idx, s, 32);
    const float ws = __shfl(wgt, s, 32);
    if (s + 1 < S) {
      int is2 = __shfl(idx, s + 1, 32);
      __builtin_prefetch(neiBase + (size_t)is2 * (D_DIM / 4) + lane, 0, 0);
    }
    const float4 v = neiBase[(size_t)is * (D_DIM / 4) + lane];
    acc.x += ws * v.x;
    acc.y += ws * v.y;
    acc.z += ws * v.z;
    acc.w += ws * v.w;
  }
  ((float4*)out)[(size_t)n * (D_DIM / 4) + lane] = acc;
}

// ---------------------------------------------------------------------------
extern "C" void kernel_launch(void* const* d_in, const int* in_sizes, int n_in,
                              void* d_out, int out_size, void* d_ws,
                              size_t ws_size, hipStream_t stream) {
  const float* h_self  = (const float*)d_in[0];
  const float* h_nei1  = (const float*)d_in[1];
  const float* h_nei2  = (const float*)d_in[2];
  const float* att1    = (const float*)d_in[3];
  const float* att2    = (const float*)d_in[4];
  const int*   nei_idx1 = (const int*)d_in[5];
  const int*   nei_idx2 = (const int*)d_in[6];
  const int*   deg1     = (const int*)d_in[7];
  const int*   deg2     = (const int*)d_in[8];

  const int N  = in_sizes[0] / D_DIM;   // 50000
  const int M1 = in_sizes[1] / D_DIM;   // 50000
  const int M2 = in_sizes[2] / D_DIM;   // 20000

  // workspace: four projection arrays (~680 KB total)
  float* projS1 = (float*)d_ws;
  float* projS2 = projS1 + N;
  float* projN1 = projS2 + N;
  float* projN2 = projN1 + M1;

  const int WPB = 8;                    // waves per block (256 threads)
  dim3 blk(32 * WPB);

  // 1) attention projections via fp32 WMMA
  dim3 g1((N  + 16 * WPB - 1) / (16 * WPB));
  dim3 g2((M1 + 16 * WPB - 1) / (16 * WPB));
  dim3 g3((M2 + 16 * WPB - 1) / (16 * WPB));
  proj_wmma_kernel<<<g1, blk, 0, stream>>>(h_self, N, att1, att2, 1,
                                           projS1, projS2);
  proj_wmma_kernel<<<g2, blk, 0, stream>>>(h_nei1, M1, att1 + D_DIM,
                                           att1 + D_DIM, 0, projN1, projN1);
  proj_wmma_kernel<<<g3, blk, 0, stream>>>(h_nei2, M2, att2 + D_DIM,
                                           att2 + D_DIM, 0, projN2, projN2);

  // 2) sample + softmax + gather-aggregate (one wave per node)
  dim3 grid((N + WPB - 1) / WPB);
  float* out1 = (float*)d_out;
  float* out2 = out1 + (size_t)N * D_DIM;
  gat_agg_kernel<<<grid, blk, 0, stream>>>(h_self, h_nei1, nei_idx1, deg1,
                                           projS1, projN1, out1, N, 8,
                                           0x1234567u);
  gat_agg_kernel<<<grid, blk, 0, stream>>>(h_self, h_nei2, nei_idx2, deg2,
                                           projS2, projN2, out2, N, 16,
                                           0x89ABCDEu);
}